// GNNPolicy_83416854823703
// MI455X (gfx1250) — compile-verified
//
#include <hip/hip_runtime.h>

typedef __attribute__((ext_vector_type(2))) float v2f;
typedef __attribute__((ext_vector_type(8))) float v8f;

#define NEG_SLOPE 0.2f

__device__ __forceinline__ unsigned fenc(float f) {
  unsigned u = __float_as_uint(f);
  return (u & 0x80000000u) ? ~u : (u | 0x80000000u);
}
__device__ __forceinline__ float fdec(unsigned u) {
  return (u & 0x80000000u) ? __uint_as_float(u & 0x7fffffffu)
                           : __uint_as_float(~u);
}

// =====================================================================
// WMMA fp32 GEMM: Y[M,64] = X[M,64] @ W[64,64] + bias[64]
// One wave computes a 16x64 tile of Y with 4 accumulators; W staged in LDS.
// =====================================================================
__global__ __launch_bounds__(256) void gemm64_bias_k(
    const float* __restrict__ X, const float* __restrict__ W,
    const float* __restrict__ bias, float* __restrict__ Y, int M) {
  __shared__ float Wl[64 * 64];
  for (int t = threadIdx.x; t < 64 * 64; t += 256) Wl[t] = W[t];
  __syncthreads();
  const int wave = threadIdx.x >> 5;
  const int lane = threadIdx.x & 31;
  const int tile = blockIdx.x * 8 + wave;
  if (tile * 16 >= M) return;
  const int row0 = tile * 16;
  const int lm = lane & 15;    // M (A,C) or N (B,C) within tile
  const int hi = lane >> 4;    // half-wave selects K pair
  v8f acc0 = {}, acc1 = {}, acc2 = {}, acc3 = {};
  const float* xrow = X + (size_t)(row0 + lm) * 64;
#pragma unroll
  for (int k0 = 0; k0 < 64; k0 += 4) {
    const int kk = k0 + 2 * hi;
    v2f a;
    a.x = xrow[kk];
    a.y = xrow[kk + 1];
    v2f b0, b1, b2, b3;
    b0.x = Wl[kk * 64 + lm];       b0.y = Wl[(kk + 1) * 64 + lm];
    b1.x = Wl[kk * 64 + 16 + lm];  b1.y = Wl[(kk + 1) * 64 + 16 + lm];
    b2.x = Wl[kk * 64 + 32 + lm];  b2.y = Wl[(kk + 1) * 64 + 32 + lm];
    b3.x = Wl[kk * 64 + 48 + lm];  b3.y = Wl[(kk + 1) * 64 + 48 + lm];
    acc0 = __builtin_amdgcn_wmma_f32_16x16x4_f32(false, a, false, b0, (short)0, acc0, false, false);
    acc1 = __builtin_amdgcn_wmma_f32_16x16x4_f32(false, a, false, b1, (short)0, acc1, false, false);
    acc2 = __builtin_amdgcn_wmma_f32_16x16x4_f32(false, a, false, b2, (short)0, acc2, false, false);
    acc3 = __builtin_amdgcn_wmma_f32_16x16x4_f32(false, a, false, b3, (short)0, acc3, false, false);
  }
#pragma unroll
  for (int r = 0; r < 8; ++r) {
    float* y = Y + (size_t)(row0 + r + 8 * hi) * 64;
    y[lm]      = acc0[r] + bias[lm];
    y[16 + lm] = acc1[r] + bias[16 + lm];
    y[32 + lm] = acc2[r] + bias[32 + lm];
    y[48 + lm] = acc3[r] + bias[48 + lm];
  }
}

// =====================================================================
// Utility kernels
// =====================================================================
__global__ void zero_k(float* __restrict__ p, long n) {
  long i = (long)blockIdx.x * blockDim.x + threadIdx.x;
  if (i < n) p[i] = 0.0f;
}

// accumulate degree and incoming edge-attr sums per destination node
__global__ void la_accum_k(const int* __restrict__ dst,
                           const float* __restrict__ ea,
                           float* __restrict__ la, float* __restrict__ deg,
                           int E) {
  int i = blockIdx.x * blockDim.x + threadIdx.x;
  if (i >= E) return;
  int d = dst[i];
  atomicAdd(&deg[d], 1.0f);
  atomicAdd(&la[2 * d],     ea[2 * i]);
  atomicAdd(&la[2 * d + 1], ea[2 * i + 1]);
}

__global__ void la_norm_k(float* __restrict__ la, const float* __restrict__ deg,
                          int Nn) {
  int n = blockIdx.x * blockDim.x + threadIdx.x;
  if (n >= Nn) return;
  float dg = fmaxf(deg[n], 1.0f);
  la[2 * n]     /= dg;
  la[2 * n + 1] /= dg;
}

// =====================================================================
// Edge kernels. Virtual edge ids [E, E+Nn) are the self loops.
// =====================================================================
template <int H, int CSHIFT>
__global__ void edge_logits_k(const float* __restrict__ xl,
                              const float* __restrict__ xr,
                              const float* __restrict__ We,   // [2,64]
                              const float* __restrict__ att,  // [64]
                              const int* __restrict__ src,
                              const int* __restrict__ dst,
                              const float* __restrict__ ea,
                              const float* __restrict__ la,
                              float* __restrict__ logits,
                              unsigned* __restrict__ mbuf, int E, int Nn) {
  int i = blockIdx.x * blockDim.x + threadIdx.x;
  if (i >= E + Nn) return;
  int s, d;
  float e0, e1;
  if (i < E) {
    s = src[i]; d = dst[i];
    e0 = ea[2 * i]; e1 = ea[2 * i + 1];
  } else {
    s = i - E; d = s;
    e0 = la[2 * s]; e1 = la[2 * s + 1];
  }
  const float4* A  = reinterpret_cast<const float4*>(xl + (size_t)s * 64);
  const float4* B  = reinterpret_cast<const float4*>(xr + (size_t)d * 64);
  const float4* W0 = reinterpret_cast<const float4*>(We);
  const float4* W1 = reinterpret_cast<const float4*>(We + 64);
  const float4* AT = reinterpret_cast<const float4*>(att);
  float lg[H];
#pragma unroll
  for (int h = 0; h < H; ++h) lg[h] = 0.0f;
#pragma unroll
  for (int j = 0; j < 16; ++j) {
    float4 a = A[j], b = B[j], w0 = W0[j], w1 = W1[j], t = AT[j];
    float v;
    v = a.x + b.x + e0 * w0.x + e1 * w1.x; v = v > 0.f ? v : NEG_SLOPE * v;
    lg[(4 * j + 0) >> CSHIFT] += v * t.x;
    v = a.y + b.y + e0 * w0.y + e1 * w1.y; v = v > 0.f ? v : NEG_SLOPE * v;
    lg[(4 * j + 1) >> CSHIFT] += v * t.y;
    v = a.z + b.z + e0 * w0.z + e1 * w1.z; v = v > 0.f ? v : NEG_SLOPE * v;
    lg[(4 * j + 2) >> CSHIFT] += v * t.z;
    v = a.w + b.w + e0 * w0.w + e1 * w1.w; v = v > 0.f ? v : NEG_SLOPE * v;
    lg[(4 * j + 3) >> CSHIFT] += v * t.w;
  }
#pragma unroll
  for (int h = 0; h < H; ++h) {
    logits[(size_t)i * H + h] = lg[h];
    atomicMax(&mbuf[(size_t)d * H + h], fenc(lg[h]));
  }
}

// exp(logit - max) in place, accumulate softmax denominator per (node, head)
template <int H>
__global__ void denom_k(float* __restrict__ logits,
                        const unsigned* __restrict__ mbuf,
                        const int* __restrict__ dst,
                        float* __restrict__ denom, int E, int Nn) {
  int i = blockIdx.x * blockDim.x + threadIdx.x;
  if (i >= E + Nn) return;
  int d = (i < E) ? dst[i] : (i - E);
#pragma unroll
  for (int h = 0; h < H; ++h) {
    float m = fdec(mbuf[(size_t)d * H + h]);
    float ex = __expf(logits[(size_t)i * H + h] - m);
    logits[(size_t)i * H + h] = ex;
    atomicAdd(&denom[(size_t)d * H + h], ex);
  }
}

// out[d] += xl[s] * alpha  (64 channels, H heads of C channels each)
template <int H, int C>
__global__ void agg_k(const float* __restrict__ xl,
                      const float* __restrict__ expl,
                      const float* __restrict__ denom,
                      const int* __restrict__ src, const int* __restrict__ dst,
                      float* __restrict__ out, int E, int Nn) {
  int i = blockIdx.x * blockDim.x + threadIdx.x;
  if (i >= E + Nn) return;
  int s, d;
  if (i < E) { s = src[i]; d = dst[i]; }
  else       { s = i - E; d = s; }
  float alpha[H];
#pragma unroll
  for (int h = 0; h < H; ++h)
    alpha[h] = expl[(size_t)i * H + h] / (denom[(size_t)d * H + h] + 1e-16f);
  const float* xs = xl + (size_t)s * 64;
  float* od = out + (size_t)d * 64;
#pragma unroll
  for (int ch = 0; ch < 64; ++ch)
    atomicAdd(&od[ch], xs[ch] * alpha[ch / C]);
}

__global__ void bias_relu_k(const float* __restrict__ in,
                            const float* __restrict__ bias,
                            float* __restrict__ out, long n) {
  long i = (long)blockIdx.x * blockDim.x + threadIdx.x;
  if (i >= n) return;
  float v = in[i] + bias[i & 63];
  out[i] = v > 0.f ? v : 0.f;
}

// y[N,8] = tanh(h[N,64] @ Wm[64,8] + bm)
__global__ void mlp_tanh_k(const float* __restrict__ h,
                           const float* __restrict__ Wm,
                           const float* __restrict__ bm, float* __restrict__ y,
                           int Nn) {
  int idx = blockIdx.x * blockDim.x + threadIdx.x;
  if (idx >= Nn * 8) return;
  int n = idx >> 3, j = idx & 7;
  const float* hr = h + (size_t)n * 64;
  float acc = bm[j];
#pragma unroll
  for (int k = 0; k < 64; ++k) acc += hr[k] * Wm[k * 8 + j];
  y[idx] = tanhf(acc);
}

// =====================================================================
// Host-side orchestration
// =====================================================================
static inline int cdiv(long a, long b) { return (int)((a + b - 1) / b); }

extern "C" void kernel_launch(void* const* d_in, const int* in_sizes, int n_in,
                              void* d_out, int out_size, void* d_ws,
                              size_t ws_size, hipStream_t stream) {
  const float* x    = (const float*)d_in[0];
  const int*   ei   = (const int*)d_in[1];
  const float* ea   = (const float*)d_in[2];
  const float* Wl1  = (const float*)d_in[3];
  const float* bl1  = (const float*)d_in[4];
  const float* Wr1  = (const float*)d_in[5];
  const float* br1  = (const float*)d_in[6];
  const float* We1  = (const float*)d_in[7];
  const float* att1 = (const float*)d_in[8];
  const float* b1   = (const float*)d_in[9];
  const float* Wl2  = (const float*)d_in[10];
  const float* bl2  = (const float*)d_in[11];
  const float* Wr2  = (const float*)d_in[12];
  const float* br2  = (const float*)d_in[13];
  const float* We2  = (const float*)d_in[14];
  const float* att2 = (const float*)d_in[15];
  const float* b2   = (const float*)d_in[16];
  const float* Wm   = (const float*)d_in[17];
  const float* bm   = (const float*)d_in[18];

  const int Nn = in_sizes[0] / 64;   // nodes
  const int E  = in_sizes[2] / 2;    // edges
  const int* src = ei;
  const int* dst = ei + E;
  const long T = (long)E + Nn;       // edges incl. self loops

  // workspace layout (floats)
  float*    xl     = (float*)d_ws;                    // 64*Nn
  float*    xr     = xl + 64L * Nn;                   // 64*Nn
  float*    agg    = xr + 64L * Nn;                   // 64*Nn
  float*    hbuf   = agg + 64L * Nn;                  // 64*Nn
  float*    logits = hbuf + 64L * Nn;                 // 4*(E+Nn)
  float*    denom  = logits + 4L * T;                 // 4*Nn
  unsigned* mbuf   = (unsigned*)(denom + 4L * Nn);    // 4*Nn
  float*    la     = (float*)(mbuf + 4L * Nn);        // 2*Nn
  float*    deg    = la + 2L * Nn;                    // Nn (contiguous w/ la)

  const int B = 256;
  const int gT   = cdiv(T, B);
  const int gE   = cdiv(E, B);
  const int gN   = cdiv(Nn, B);
  const int g64N = cdiv(64L * Nn, B);
  const int gGemm = cdiv(Nn, 128);  // 8 waves * 16 rows per block

  // ---- self-loop edge attr: mean of incoming edge attrs ----
  zero_k<<<cdiv(3L * Nn, B), B, 0, stream>>>(la, 3L * Nn);  // la + deg
  la_accum_k<<<gE, B, 0, stream>>>(dst, ea, la, deg, E);
  la_norm_k<<<gN, B, 0, stream>>>(la, deg, Nn);

  // ---- layer 1 (H=4, C=16) ----
  gemm64_bias_k<<<gGemm, B, 0, stream>>>(x, Wl1, bl1, xl, Nn);
  gemm64_bias_k<<<gGemm, B, 0, stream>>>(x, Wr1, br1, xr, Nn);
  zero_k<<<cdiv(4L * Nn, B), B, 0, stream>>>(denom, 4L * Nn);
  zero_k<<<cdiv(4L * Nn, B), B, 0, stream>>>((float*)mbuf, 4L * Nn);
  edge_logits_k<4, 4><<<gT, B, 0, stream>>>(xl, xr, We1, att1, src, dst, ea,
                                            la, logits, mbuf, E, Nn);
  denom_k<4><<<gT, B, 0, stream>>>(logits, mbuf, dst, denom, E, Nn);
  zero_k<<<g64N, B, 0, stream>>>(agg, 64L * Nn);
  agg_k<4, 16><<<gT, B, 0, stream>>>(xl, logits, denom, src, dst, agg, E, Nn);
  bias_relu_k<<<g64N, B, 0, stream>>>(agg, b1, hbuf, 64L * Nn);

  // ---- layer 2 (H=2, C=32) ----
  gemm64_bias_k<<<gGemm, B, 0, stream>>>(hbuf, Wl2, bl2, xl, Nn);
  gemm64_bias_k<<<gGemm, B, 0, stream>>>(hbuf, Wr2, br2, xr, Nn);
  zero_k<<<cdiv(4L * Nn, B), B, 0, stream>>>(denom, 4L * Nn);
  zero_k<<<cdiv(4L * Nn, B), B, 0, stream>>>((float*)mbuf, 4L * Nn);
  edge_logits_k<2, 5><<<gT, B, 0, stream>>>(xl, xr, We2, att2, src, dst, ea,
                                            la, logits, mbuf, E, Nn);
  denom_k<2><<<gT, B, 0, stream>>>(logits, mbuf, dst, denom, E, Nn);
  zero_k<<<g64N, B, 0, stream>>>(agg, 64L * Nn);
  agg_k<2, 32><<<gT, B, 0, stream>>>(xl, logits, denom, src, dst, agg, E, Nn);
  bias_relu_k<<<g64N, B, 0, stream>>>(agg, b2, hbuf, 64L * Nn);

  // ---- final MLP + tanh ----
  mlp_tanh_k<<<cdiv(8L * Nn, B), B, 0, stream>>>(hbuf, Wm, bm, (float*)d_out,
                                                 Nn);
}